// FullSpectralAttentionLayer_24833500906131
// MI455X (gfx1250) — compile-verified
//
#include <hip/hip_runtime.h>
#include <hip/hip_bf16.h>
#include <math.h>

// Problem constants
#define N_BATCH 32
#define CHANNEL 256
#define HW      4096      // 64*64
#define NC      8192      // N_BATCH*CHANNEL
#define CR      16        // CHANNEL / REDUCTION

typedef __attribute__((ext_vector_type(2))) float v2f;
typedef __attribute__((ext_vector_type(8))) float v8f;

// ---------------------------------------------------------------------------
// Kernel 0: build the folded DCT kernel G[h][w] = (1/64) * sum_l bh[l,h]*bh[l,w]
// (bh == bw since H == W == L == 64). Double precision, cast to f32.
// ---------------------------------------------------------------------------
__global__ void fsa_compute_G(float* __restrict__ G) {
    int idx = blockIdx.x * blockDim.x + threadIdx.x;   // 0..4095
    int h = idx >> 6;
    int w = idx & 63;
    const double PI = 3.14159265358979323846;
    double sum = 0.0;
    for (int l = 0; l < 64; ++l) {
        double sc = (l == 0) ? 1.0 : 1.4142135623730951; // sqrt(2) for l>0
        double bh = cos(PI * (double)l * ((double)h + 0.5) / 64.0) * 0.125 * sc;
        double bw = cos(PI * (double)l * ((double)w + 0.5) / 64.0) * 0.125 * sc;
        sum += bh * bw;
    }
    G[idx] = (float)(sum / 64.0);   // mean over the 64 diagonal filters
}

// ---------------------------------------------------------------------------
// Kernel 1: y[n,c] = <x plane, G>. One block per (n,c) plane.
// 256 threads * 4 float4 = 4096 elements. G is 16KB -> L2 resident.
// ---------------------------------------------------------------------------
__global__ void fsa_reduce_y(const float* __restrict__ x,
                             const float* __restrict__ G,
                             float* __restrict__ y) {
    __shared__ float red[256];
    const int t = threadIdx.x;
    const float4* p4 = (const float4*)(x + (size_t)blockIdx.x * HW);
    const float4* g4 = (const float4*)G;
    float acc = 0.0f;
#pragma unroll
    for (int i = 0; i < 4; ++i) {
        float4 a = p4[t + 256 * i];
        float4 g = g4[t + 256 * i];
        acc += a.x * g.x + a.y * g.y + a.z * g.z + a.w * g.w;
    }
    red[t] = acc;
    __syncthreads();
    for (int off = 128; off > 0; off >>= 1) {
        if (t < off) red[t] += red[t + off];
        __syncthreads();
    }
    if (t == 0) y[blockIdx.x] = red[0];
}

// ---------------------------------------------------------------------------
// Kernel 2: the MLP on a single wave32 using V_WMMA_F32_16X16X4_F32.
//   h = relu(y @ w1^T)        (M=32, N=16,  K=256)
//   s = sigmoid(h @ w2^T)     (M=32, N=256, K=16)
// A layout (16x4 f32): lanes 0-15 -> M=lane, VGPR0=K0,VGPR1=K1;
//                      lanes 16-31 -> M=lane-16, VGPR0=K2,VGPR1=K3.
// B layout (4x16 f32): lanes 0-15 -> N=lane, VGPR0=K0,VGPR1=K1;
//                      lanes 16-31 -> N=lane-16, VGPR0=K2,VGPR1=K3.
// C/D layout: VGPR i -> lanes 0-15: (M=i, N=lane); lanes 16-31: (M=i+8, N=lane-16).
// ---------------------------------------------------------------------------
__global__ void fsa_mlp_wmma(const float* __restrict__ y,   // [32,256]
                             const float* __restrict__ w1,  // [16,256]
                             const float* __restrict__ w2,  // [256,16]
                             float* __restrict__ s) {       // [32,256]
    __shared__ float lds_h[32][CR];   // relu(y @ w1^T)

    const int lane = threadIdx.x;     // 0..31, single wave, EXEC all-ones
    const int half = lane >> 4;       // 0 or 1
    const int l16  = lane & 15;

    // ---- GEMM 1: h[n,j] = relu(sum_c y[n,c] * w1[j,c]) ----
    for (int mt = 0; mt < 2; ++mt) {
        v8f acc = {};
        const float* yrow = y + (size_t)(mt * 16 + l16) * CHANNEL;  // A row M=l16
        const float* brow = w1 + (size_t)l16 * CHANNEL;             // B col N=l16
        for (int k0 = 0; k0 < CHANNEL; k0 += 4) {
            const int k = k0 + 2 * half;
            v2f a, b;
            a.x = yrow[k + 0];
            a.y = yrow[k + 1];
            b.x = brow[k + 0];   // B[k, n=l16] = w1[l16, k]
            b.y = brow[k + 1];
            acc = __builtin_amdgcn_wmma_f32_16x16x4_f32(
                false, a, false, b, (short)0, acc, false, false);
        }
#pragma unroll
        for (int i = 0; i < 8; ++i) {
            float v = acc[i];
            lds_h[mt * 16 + i + 8 * half][l16] = (v > 0.0f) ? v : 0.0f;
        }
    }
    __syncthreads();

    // ---- GEMM 2: s[n,c] = sigmoid(sum_j h[n,j] * w2[c,j]) ----
    for (int mt = 0; mt < 2; ++mt) {
        for (int ct = 0; ct < 16; ++ct) {
            v8f acc = {};
            const int c = ct * 16 + l16;          // B col N = c
            for (int k0 = 0; k0 < CR; k0 += 4) {
                const int k = k0 + 2 * half;
                v2f a, b;
                a.x = lds_h[mt * 16 + l16][k + 0];  // A row M=l16
                a.y = lds_h[mt * 16 + l16][k + 1];
                b.x = w2[(size_t)c * CR + k + 0];   // B[k, c] = w2[c, k]
                b.y = w2[(size_t)c * CR + k + 1];
                acc = __builtin_amdgcn_wmma_f32_16x16x4_f32(
                    false, a, false, b, (short)0, acc, false, false);
            }
#pragma unroll
            for (int i = 0; i < 8; ++i) {
                const int n = mt * 16 + i + 8 * half;
                float v = acc[i];
                s[(size_t)n * CHANNEL + c] = 1.0f / (1.0f + expf(-v));
            }
        }
    }
}

// ---------------------------------------------------------------------------
// Kernel 3: out[n,c,:,:] = x[n,c,:,:] * s[n,c].  One block per plane, float4.
// ---------------------------------------------------------------------------
__global__ void fsa_scale_x(const float* __restrict__ x,
                            const float* __restrict__ s,
                            float* __restrict__ out) {
    const int t = threadIdx.x;
    const size_t base = (size_t)blockIdx.x * (HW / 4);  // in float4 units
    const float4* p4 = (const float4*)x;
    float4* o4 = (float4*)out;
    const float sc = s[blockIdx.x];
#pragma unroll
    for (int i = 0; i < 4; ++i) {
        float4 a = p4[base + t + 256 * i];
        a.x *= sc; a.y *= sc; a.z *= sc; a.w *= sc;
        o4[base + t + 256 * i] = a;
    }
}

// ---------------------------------------------------------------------------
// Launch
// ---------------------------------------------------------------------------
extern "C" void kernel_launch(void* const* d_in, const int* in_sizes, int n_in,
                              void* d_out, int out_size, void* d_ws, size_t ws_size,
                              hipStream_t stream) {
    const float* x  = (const float*)d_in[0];   // [32,256,64,64]
    const float* w1 = (const float*)d_in[1];   // [16,256]
    const float* w2 = (const float*)d_in[2];   // [256,16]
    float* out = (float*)d_out;                // [32,256,64,64]

    float* G = (float*)d_ws;       // 4096 floats
    float* y = G + HW;             // 8192 floats
    float* s = y + NC;             // 8192 floats

    fsa_compute_G<<<16, 256, 0, stream>>>(G);
    fsa_reduce_y<<<NC, 256, 0, stream>>>(x, G, y);
    fsa_mlp_wmma<<<1, 32, 0, stream>>>(y, w1, w2, s);
    fsa_scale_x<<<NC, 256, 0, stream>>>(x, s, out);
}